// GlassBlur_59906203844652
// MI455X (gfx1250) — compile-verified
//
#include <hip/hip_runtime.h>

#define HH 384
#define WW 384
#define NT 24          // 16x16 tiles per dimension
#define AP 392         // padded A row: 4 zero cols + 384 + 4 zero cols
#define SCAN_R 40      // LDS row ring (40*384*4 = 60 KB)

typedef float v2f __attribute__((ext_vector_type(2)));
typedef float v8f __attribute__((ext_vector_type(8)));

// ---------------------------------------------------------------------------
// Gaussian kernel (sigma=0.4, radius=2, normalized).
// ---------------------------------------------------------------------------
__device__ __forceinline__ void gauss_k5(float k[5]) {
    float s = 0.f;
#pragma unroll
    for (int d = -2; d <= 2; ++d) {
        float x = (float)d / 0.4f;
        float v = __expf(-0.5f * x * x);
        k[d + 2] = v;
        s += v;
    }
    float inv = 1.f / s;
#pragma unroll
    for (int i = 0; i < 5; ++i) k[i] *= inv;
}

// ---------------------------------------------------------------------------
// Build padded banded blur matrix:  Apad[i][j+4] = A[i][j], zero border.
// A[i][j]: out[i] = sum_d k[d+2]*img[clamp(i+d)] with 'edge' padding lumped.
// Built once per launch; reused by all four GEMM passes (L2-resident).
// ---------------------------------------------------------------------------
__global__ __launch_bounds__(256) void gb_build_A(float* __restrict__ Apad) {
    int idx = blockIdx.x * 256 + threadIdx.x;
    if (idx >= HH * AP) return;
    int i = idx / AP, j = idx % AP - 4;
    float k5[5];
    gauss_k5(k5);
    float v = 0.f;
    if (j >= 0 && j < HH) {
#pragma unroll
        for (int d = -2; d <= 2; ++d) {
            int s = i + d;
            s = s < 0 ? 0 : (s > HH - 1 ? HH - 1 : s);
            v += (s == j) ? k5[d + 2] : 0.f;
        }
    }
    Apad[idx] = v;
}

// ---------------------------------------------------------------------------
// T = A * X  (blur along axis 0).  Band => 6 WMMA f32 16x16x4 per 16x16 tile.
// Branchless: A-fragment pair is one aligned b64 load from the padded row.
// ---------------------------------------------------------------------------
__global__ __launch_bounds__(256) void gb_blur_rows(const float* __restrict__ X,
                                                    const float* __restrict__ Apad,
                                                    float* __restrict__ T) {
    int lane = threadIdx.x & 31;
    int tile = blockIdx.x * 8 + (threadIdx.x >> 5);
    int mi = tile / NT, ni = tile % NT;

    int m     = 16 * mi + (lane & 15);        // A-frag row
    int n     = 16 * ni + (lane & 15);        // B-frag col
    int khalf = (lane >> 4) * 2;              // K sub-index per lane half
    const float* Arow = Apad + m * AP + 4;    // Arow[k] valid for k in [-4, 387]

    v8f acc = {0.f, 0.f, 0.f, 0.f, 0.f, 0.f, 0.f, 0.f};
#pragma unroll
    for (int c = 0; c < 6; ++c) {
        int k0 = 16 * mi - 4 + 4 * c + khalf; // even; band K in [16mi-4, 16mi+19]
        v2f a = *(const v2f*)(Arow + k0);     // global_load_b64, always in padded bounds
        int kc0 = min(max(k0, 0), HH - 1);    // clamped rows; a==0 outside band
        int kc1 = min(max(k0 + 1, 0), HH - 1);
        v2f b;
        b.x = X[kc0 * WW + n];
        b.y = X[kc1 * WW + n];
        acc = __builtin_amdgcn_wmma_f32_16x16x4_f32(false, a, false, b,
                                                    (short)0, acc, false, false);
    }
    int col   = 16 * ni + (lane & 15);
    int rbase = 16 * mi + (lane >> 4) * 8;    // C/D layout: vgpr r -> M=r (+8 hi half)
#pragma unroll
    for (int r = 0; r < 8; ++r) T[(rbase + r) * WW + col] = acc[r];
}

// ---------------------------------------------------------------------------
// Y = T * A^T  (blur along axis 1), optional clip to [0,1] on the final pass.
// B[k][n] = A[16ni+n][k] -> b64 from padded row; A-frag is clamped-base b64
// from T (garbage lanes are multiplied by B==0).
// ---------------------------------------------------------------------------
__global__ __launch_bounds__(256) void gb_blur_cols(const float* __restrict__ T,
                                                    const float* __restrict__ Apad,
                                                    float* __restrict__ Y,
                                                    int clip) {
    int lane = threadIdx.x & 31;
    int tile = blockIdx.x * 8 + (threadIdx.x >> 5);
    int mi = tile / NT, ni = tile % NT;

    int m     = 16 * mi + (lane & 15);
    int jn    = 16 * ni + (lane & 15);
    int khalf = (lane >> 4) * 2;
    const float* Arow = Apad + jn * AP + 4;

    v8f acc = {0.f, 0.f, 0.f, 0.f, 0.f, 0.f, 0.f, 0.f};
#pragma unroll
    for (int c = 0; c < 6; ++c) {
        int k0 = 16 * ni - 4 + 4 * c + khalf; // even
        v2f b = *(const v2f*)(Arow + k0);     // b64
        int kcb = min(max(k0, 0), WW - 2);    // even -> 8B-aligned b64 from T
        v2f a = *(const v2f*)(T + m * WW + kcb);
        acc = __builtin_amdgcn_wmma_f32_16x16x4_f32(false, a, false, b,
                                                    (short)0, acc, false, false);
    }
    int col   = 16 * ni + (lane & 15);
    int rbase = 16 * mi + (lane >> 4) * 8;
#pragma unroll
    for (int r = 0; r < 8; ++r) {
        float v = acc[r];
        if (clip) v = fminf(fmaxf(v, 0.f), 1.f);
        Y[(rbase + r) * WW + col] = v;
    }
}

// ---------------------------------------------------------------------------
// Sequential swap scan as a 32-lane skewed pipeline in LDS.
// Lane L executes pass h = hb - L delayed by 2L column-steps; step (h,w)
// depends only on (h,w+1) [same lane, prev iter] and (h+1,w-1..w+1) [lane
// L-1, earlier iters]; concurrent lanes touch disjoint cells.  40-row LDS
// ring streams rows against global; offsets are prefetched per block.
// ---------------------------------------------------------------------------
__global__ __launch_bounds__(32) void gb_glass_scan(float* __restrict__ img,
                                                    const int* __restrict__ off) {
    __shared__ float lds[SCAN_R][WW];
    int lane = threadIdx.x;
    const int nOff = (HH - 2) * (WW - 2) * 2;

    // initial window: rows 383..351
    for (int r = HH - 1; r >= HH - 33; --r)
        for (int c = lane; c < WW; c += 32)
            lds[r % SCAN_R][c] = img[r * WW + c];
    asm volatile("s_wait_loadcnt 0x0\n\ts_wait_dscnt 0x0" ::: "memory");

    for (int blk = 0; blk < 12; ++blk) {
        int hb = (HH - 1) - 32 * blk;
        int h  = hb - lane;                       // this lane's pass row
        int hs   = (h >= 0 ? h : 0) % SCAN_R;
        int hm1s = (h >= 1 ? h - 1 : 0) % SCAN_R;

        // prefetch this block's sequential offset stream (~96 KB) into cache
        int obase = (32 * blk) * (WW - 2) * 2;
        for (int i = lane * 32; i < 32 * (WW - 2) * 2 && obase + i < nOff; i += 32 * 32)
            __builtin_prefetch(&off[obase + i], 0, 1);   // global_prefetch_b8

        for (int t = 0; t < (WW - 2) + 62; ++t) { // skewed wavefront
            int s = t - 2 * lane;
            if (h >= 2 && s >= 0 && s < (WW - 2)) {
                int w   = (WW - 1) - s;
                int idx = ((HH - 1) - h) * (WW - 2) + s;
                int dx  = off[2 * idx + 0];
                int dy  = off[2 * idx + 1];
                int wp  = w + dx;
                int rps = (dy == 0) ? hs : hm1s;
                float a = lds[hs][w];
                float b = lds[rps][wp];
                lds[hs][w]   = b;
                lds[rps][wp] = a;                 // (0,0) offset restores a
            }
            asm volatile("s_wait_dscnt 0x0" ::: "memory");
        }

        // rows hb..hb-31 have had both their passes -> final, write out
        for (int r = hb; r >= hb - 31 && r >= 0; --r)
            for (int c = lane; c < WW; c += 32)
                img[r * WW + c] = lds[r % SCAN_R][c];
        // load next window rows hb-33..hb-64 into now-dead ring slots
        for (int r = hb - 33; r >= hb - 64 && r >= 0; --r)
            for (int c = lane; c < WW; c += 32)
                lds[r % SCAN_R][c] = img[r * WW + c];
        asm volatile("s_wait_loadcnt 0x0\n\ts_wait_storecnt 0x0\n\ts_wait_dscnt 0x0" ::: "memory");
    }
}

// ---------------------------------------------------------------------------
extern "C" void kernel_launch(void* const* d_in, const int* in_sizes, int n_in,
                              void* d_out, int out_size, void* d_ws, size_t ws_size,
                              hipStream_t stream) {
    const float* img     = (const float*)d_in[0];
    const int*   offsets = (const int*)d_in[1];
    float* Apad = (float*)d_ws;         // 384*392 f32  (602 KB)
    float* buf  = Apad + HH * AP;       // working image (576 KB)
    float* tmp  = buf + HH * WW;        // GEMM intermediate (576 KB)
    float* out  = (float*)d_out;

    gb_build_A<<<(HH * AP + 255) / 256, 256, 0, stream>>>(Apad);

    dim3 blk(256), grd(72);             // 72*8 waves = 576 tiles = 24*24
    gb_blur_rows<<<grd, blk, 0, stream>>>(img, Apad, tmp);
    gb_blur_cols<<<grd, blk, 0, stream>>>(tmp, Apad, buf, 0);
    gb_glass_scan<<<1, 32, 0, stream>>>(buf, offsets);
    gb_blur_rows<<<grd, blk, 0, stream>>>(buf, Apad, tmp);
    gb_blur_cols<<<grd, blk, 0, stream>>>(tmp, Apad, out, 1);
}